// TensorizedAEloss_29308856828175
// MI455X (gfx1250) — compile-verified
//
#include <hip/hip_runtime.h>
#include <stdint.h>

typedef __attribute__((ext_vector_type(2))) float v2f;
typedef __attribute__((ext_vector_type(8))) float v8f;

#define NT   16      // tensors
#define NB   8192    // batch
#define NF   512     // in_feature
#define NE   64      // emb_dim
#define FC   128     // features per LDS chunk
#define NCH  32      // K=4 WMMA chunks per feature chunk
#define NFC  4       // feature chunks (512/128)
#define LSTR 132     // padded LDS row stride (floats): rows hit distinct banks
#define REGP 32      // partial blocks per tensor for reg reduction

// ---------------- reg_term reduction (deterministic, no atomics) -------------
__global__ __launch_bounds__(256)
void reg_partial_kernel(const float* __restrict__ emb, float* __restrict__ ws) {
    const int blk = blockIdx.x;                    // 0..NT*REGP-1
    const int per_block = NB * NE / REGP;          // 16384 elements
    const float* base = emb + (size_t)blk * per_block;
    float s = 0.f;
    for (int i = threadIdx.x; i < per_block; i += 256) {
        float v = base[i];
        s = fmaf(v, v, s);
    }
    __shared__ float red[256];
    red[threadIdx.x] = s;
    __syncthreads();
    for (int w = 128; w >= 1; w >>= 1) {
        if (threadIdx.x < w) red[threadIdx.x] += red[threadIdx.x + w];
        __syncthreads();
    }
    if (threadIdx.x == 0) ws[NT + blk] = red[0];
}

__global__ __launch_bounds__(32)
void reg_final_kernel(float* __restrict__ ws) {
    const int t = threadIdx.x;
    if (t < NT) {
        float s = 0.f;
        for (int p = 0; p < REGP; ++p) s += ws[NT + t * REGP + p];  // fixed order
        ws[t] = 0.001f * s;
    }
}

// -------- gfx1250 async copy: one 16-row x 128-float tile, 16 wave-wide b128 --
__device__ __forceinline__ void async_row_tile(const float* __restrict__ gsrc,
                                               const float* lds_base,
                                               int b0, int f0, int lane) {
#pragma unroll
    for (int j = 0; j < 16; ++j) {
        uint32_t ldsa = (uint32_t)(uintptr_t)(lds_base + j * LSTR + lane * 4);
        uint32_t voff = ((uint32_t)(b0 + j) * NF + (uint32_t)f0 + (uint32_t)lane * 4u) * 4u;
        // GVS mode: mem = SADDR(64b sgpr pair) + VADDR(32b) ; LDS dst byte addr in %0
        asm volatile("global_load_async_to_lds_b128 %0, %1, %2"
                     :: "v"(ldsa), "v"(voff), "s"(gsrc)
                     : "memory");
    }
}

// ---------------------------- main streaming kernel --------------------------
// grid: NB/16 blocks of 128 threads (4 wave32). Block owns 16 batch rows.
// Wave w owns tensors 4w..4w+3. mse via diag(D*D^T) with WMMA f32 16x16x4,
// passing the same register pair as A and B (symmetric operand layouts).
__global__ __launch_bounds__(128)
void loss_kernel(const float* __restrict__ rec, const float* __restrict__ xout,
                 const float* __restrict__ cent, const float* __restrict__ ws,
                 float* __restrict__ out, int out_elems) {
    __shared__ float xs[16 * LSTR];        // x_out tile chunk   (8.25 KB)
    __shared__ float cs[NT * LSTR];        // all centers chunk  (8.25 KB)
    __shared__ float rs[4][16 * LSTR];     // per-wave rec tile  (33 KB)
    __shared__ float fin[NT][16];          // per-(t,row) mse sums (1 KB)

    const int tid  = threadIdx.x;
    const int lane = tid & 31;
    const int wv   = tid >> 5;
    const int b0   = blockIdx.x * 16;
    const int row  = lane & 15;            // A-layout: lanes 0-15 and 16-31 both map M=0..15
    const int koff = (lane >> 4) << 1;     // lanes>=16 supply K=2,3 of each K=4 chunk

    float msum[4] = {0.f, 0.f, 0.f, 0.f};
    float* rsw = &rs[wv][0];

    for (int fc = 0; fc < NFC; ++fc) {
        const int f0 = fc * FC;
        // cooperatively stage x tile chunk and all-16-centers chunk (b128 loads)
#pragma unroll
        for (int i = 0; i < 4; ++i) {
            int q = tid + 128 * i;                  // 0..511 float4 slots
            int r = q >> 5, c4 = (q & 31) * 4;
            float4 v = *(const float4*)(xout + (size_t)(b0 + r) * NF + f0 + c4);
            *(float4*)(xs + r * LSTR + c4) = v;
        }
#pragma unroll
        for (int i = 0; i < 4; ++i) {
            int q = tid + 128 * i;
            int r = q >> 5, c4 = (q & 31) * 4;
            float4 v = *(const float4*)(cent + (size_t)r * NF + f0 + c4);
            *(float4*)(cs + r * LSTR + c4) = v;
        }
        __syncthreads();

#pragma unroll
        for (int tt = 0; tt < 4; ++tt) {
            const int t = wv * 4 + tt;
            async_row_tile(rec + (size_t)t * NB * NF, rsw, b0, f0, lane);
            asm volatile("s_wait_asynccnt 0x0" ::: "memory");

            v8f acc = {};
#pragma unroll
            for (int k = 0; k < NCH; ++k) {
                const int fl = 4 * k + koff;
                v2f rv = *(const v2f*)(rsw + row * LSTR + fl);
                v2f xv = *(const v2f*)(xs  + row * LSTR + fl);
                v2f cv = *(const v2f*)(cs  + t   * LSTR + fl);
                v2f a  = rv - xv + cv;                  // diff = rec - x + c
                // Gram accumulate: D += a * a^T  (A and B are the same regs)
                acc = __builtin_amdgcn_wmma_f32_16x16x4_f32(
                          false, a, false, a, (short)0, acc, false, false);
            }
            // diagonal (k,k): lanes 0-7 -> acc[lane]; lanes 24-31 -> acc[lane-24]
            int li = ((lane < 8) ? lane : (lane - 24)) & 7;
            float x0 = (li & 1) ? acc[1] : acc[0];
            float x1 = (li & 1) ? acc[3] : acc[2];
            float x2 = (li & 1) ? acc[5] : acc[4];
            float x3 = (li & 1) ? acc[7] : acc[6];
            float y0 = (li & 2) ? x1 : x0;
            float y1 = (li & 2) ? x3 : x2;
            msum[tt] += (li & 4) ? y1 : y0;
        }
        __syncthreads();   // xs/cs about to be restaged
    }

    // publish per-(t,row) sums: lanes 0-7 own rows 0-7, lanes 24-31 own rows 8-15
    {
        int r = (lane < 8) ? lane : (lane - 16);
        bool valid = (lane < 8) || (lane >= 24);
#pragma unroll
        for (int tt = 0; tt < 4; ++tt)
            if (valid) fin[wv * 4 + tt][r] = msum[tt];
    }
    __syncthreads();

    if (wv == 0 && lane < 16) {
        float tot = 0.f, best = 3.4e38f;
        int bidx = 0;
#pragma unroll
        for (int t = 0; t < NT; ++t) {
            float v = fin[t][lane] * (1.0f / 512.0f) + ws[t];  // mse/F + reg[t]
            tot += v;
            if (v < best) { best = v; bidx = t; }              // strict <, first min wins
        }
        out[b0 + lane] = tot;
        if (out_elems >= 24576) {       // int64 indices packed after 8192 f32 losses
            long long* ip = (long long*)(out + NB);
            ip[b0 + lane] = (long long)bidx;
        } else {                        // harness flattened everything to f32
            out[NB + b0 + lane] = (float)bidx;
        }
    }
}

// ------------------------------- host launcher -------------------------------
extern "C" void kernel_launch(void* const* d_in, const int* in_sizes, int n_in,
                              void* d_out, int out_size, void* d_ws, size_t ws_size,
                              hipStream_t stream) {
    (void)in_sizes; (void)n_in; (void)ws_size;
    const float* emb  = (const float*)d_in[0];
    const float* rec  = (const float*)d_in[1];
    const float* xout = (const float*)d_in[2];
    const float* cent = (const float*)d_in[3];
    float* out = (float*)d_out;
    float* ws  = (float*)d_ws;

    reg_partial_kernel<<<NT * REGP, 256, 0, stream>>>(emb, ws);
    reg_final_kernel<<<1, 32, 0, stream>>>(ws);
    loss_kernel<<<NB / 16, 128, 0, stream>>>(rec, xout, cent, ws, out, out_size);
}